// RNN_395136991900
// MI455X (gfx1250) — compile-verified
//
#include <hip/hip_runtime.h>
#include <cmath>

// RNN: B=1024, T=512, I=22, H=32, 2 layers, fused persistent-per-tile kernel.
// Transposed formulation S' = tanh(W_hh @ S + W_ih @ x^T + b); weights live in
// V_WMMA_F32_16X16X4_F32 A-fragments in registers; 16-batch-column state tile in LDS.
// Software-pipelined: iteration t runs layer1(t) and layer0(t+1) as independent
// WMMA chains in one scheduling region (~6 chains in flight).

#define B_ 1024
#define T_ 512
#define I_ 22
#define H_ 32

typedef __attribute__((ext_vector_type(2))) float v2f;
typedef __attribute__((ext_vector_type(4))) float v4f;
typedef __attribute__((ext_vector_type(8))) float v8f;

__device__ __forceinline__ v8f wmma_f32(v2f a, v2f b, v8f c) {
    // 8 args: (neg_a, A, neg_b, B, c_mod, C, reuse_a, reuse_b)
    return __builtin_amdgcn_wmma_f32_16x16x4_f32(false, a, false, b, (short)0, c, false, false);
}

#if defined(__has_builtin)
#if __has_builtin(__builtin_amdgcn_tanhf)
#define HAVE_HW_TANH 1
#endif
#endif

__device__ __forceinline__ float fast_tanh(float v) {
#ifdef HAVE_HW_TANH
    return __builtin_amdgcn_tanhf(v);   // v_tanh_f32: single TRANS op on the critical path
#else
    return tanhf(v);
#endif
}

__global__ __launch_bounds__(32) void rnn_fused_kernel(
    const float* __restrict__ x,        // [B,T,I]
    const float* __restrict__ h_in,     // [2,B,H]
    const float* __restrict__ Wih0,     // [H,I]
    const float* __restrict__ Whh0,     // [H,H]
    const float* __restrict__ bih0,
    const float* __restrict__ bhh0,
    const float* __restrict__ Wih1,    // [H,H]
    const float* __restrict__ Whh1,    // [H,H]
    const float* __restrict__ bih1,
    const float* __restrict__ bhh1,
    const float* __restrict__ Wout,    // [1,H]
    const float* __restrict__ bout,    // [1]
    float* __restrict__ r_out,         // [B,T,H]
    float* __restrict__ outs,          // [B]
    float* __restrict__ hstate_out)    // [2,B,H]
{
    __shared__ float lds[2][H_][16];    // state S[layer][k(hidden)][n(batch col)]
    __shared__ float plds[32];

    const int lane = threadIdx.x & 31;
    const int n    = lane & 15;         // batch column within tile
    const int half = lane >> 4;         // lane half (selects k/m sub-ranges)
    const int b0   = blockIdx.x * 16;   // batch base for this wave

    // ---- A-fragments (weights), resident for the whole time loop ----
    // A[m][k]: lane = (m&15), half selects k pair, vgpr j = k&1; k = 4*kk + 2*half + j
    v2f aWih0[2][6];                    // K padded 22 -> 24 (zeros)
    v2f aWhh0[2][8], aWih1[2][8], aWhh1[2][8];
#pragma unroll
    for (int mt = 0; mt < 2; ++mt) {
        const int m = 16 * mt + n;
#pragma unroll
        for (int kk = 0; kk < 6; ++kk)
#pragma unroll
            for (int j = 0; j < 2; ++j) {
                const int k = 4 * kk + 2 * half + j;
                aWih0[mt][kk][j] = (k < I_) ? Wih0[m * I_ + k] : 0.0f;
            }
#pragma unroll
        for (int kk = 0; kk < 8; ++kk)
#pragma unroll
            for (int j = 0; j < 2; ++j) {
                const int k = 4 * kk + 2 * half + j;
                aWhh0[mt][kk][j] = Whh0[m * H_ + k];
                aWih1[mt][kk][j] = Wih1[m * H_ + k];
                aWhh1[mt][kk][j] = Whh1[m * H_ + k];
            }
    }

    // ---- bias in C/D register order: elem r <-> m = 16*mt + 8*half + r ----
    float bc0[2][8], bc1[2][8];
#pragma unroll
    for (int mt = 0; mt < 2; ++mt)
#pragma unroll
        for (int r = 0; r < 8; ++r) {
            const int m = 16 * mt + 8 * half + r;
            bc0[mt][r] = bih0[m] + bhh0[m];
            bc1[mt][r] = bih1[m] + bhh1[m];
        }

    // ---- initial state into LDS (transposed): lds[l][k][n] = h_in[l][b0+n][k] ----
#pragma unroll
    for (int l = 0; l < 2; ++l)
#pragma unroll
        for (int i = 0; i < 16; ++i) {
            const int k = half + 2 * i;   // covers 0..31 across the two halves
            lds[l][k][n] = h_in[(size_t)l * B_ * H_ + (size_t)(b0 + n) * H_ + k];
        }
    // single wave per block: same-wave DS ordering makes stores visible to later loads

    const float* xr = x + (size_t)(b0 + n) * T_ * I_;

    // ================= prologue: layer0 consumes x_0 -> S0^1 =======================
    {
        v8f c0[2];
#pragma unroll
        for (int mt = 0; mt < 2; ++mt)
#pragma unroll
            for (int r = 0; r < 8; ++r) c0[mt][r] = bc0[mt][r];
#pragma unroll
        for (int kk = 0; kk < 6; ++kk) {
            const int k0 = 4 * kk + 2 * half;
            v2f b;                               // clamped index: no exec divergence,
            b[0] = xr[k0 < I_ ? k0 : I_ - 1];    // A-fragment is zero for k >= 22
            b[1] = xr[(k0 + 1) < I_ ? (k0 + 1) : I_ - 1];
            c0[0] = wmma_f32(aWih0[0][kk], b, c0[0]);
            c0[1] = wmma_f32(aWih0[1][kk], b, c0[1]);
        }
#pragma unroll
        for (int kk = 0; kk < 8; ++kk) {
            const int k0 = 4 * kk + 2 * half;
            v2f b;
            b[0] = lds[0][k0][n];
            b[1] = lds[0][k0 + 1][n];
            c0[0] = wmma_f32(aWhh0[0][kk], b, c0[0]);
            c0[1] = wmma_f32(aWhh0[1][kk], b, c0[1]);
        }
#pragma unroll
        for (int mt = 0; mt < 2; ++mt)
#pragma unroll
            for (int r = 0; r < 8; ++r)
                lds[0][16 * mt + 8 * half + r][n] = fast_tanh(c0[mt][r]);
    }

    // ================= pipelined main loop =========================================
    for (int t = 0; t < T_; ++t) {
        const bool last = (t == T_ - 1);
        const float* xt = xr + (size_t)(last ? t : t + 1) * I_;  // clamped row; dead when last

        v8f c1i[2], c1h[2], c0[2];
#pragma unroll
        for (int mt = 0; mt < 2; ++mt)
#pragma unroll
            for (int r = 0; r < 8; ++r) {
                c1i[mt][r] = bc1[mt][r];
                c1h[mt][r] = 0.0f;
                c0[mt][r]  = bc0[mt][r];
            }

        // one branchless region: 6 independent accumulation chains
#pragma unroll
        for (int kk = 0; kk < 8; ++kk) {
            const int k0 = 4 * kk + 2 * half;
            v2f ba, bb;
            ba[0] = lds[0][k0][n];  ba[1] = lds[0][k0 + 1][n];   // S0^{t+1}
            bb[0] = lds[1][k0][n];  bb[1] = lds[1][k0 + 1][n];   // S1^{t}
            c1i[0] = wmma_f32(aWih1[0][kk], ba, c1i[0]);
            c1i[1] = wmma_f32(aWih1[1][kk], ba, c1i[1]);
            c1h[0] = wmma_f32(aWhh1[0][kk], bb, c1h[0]);
            c1h[1] = wmma_f32(aWhh1[1][kk], bb, c1h[1]);
            c0[0]  = wmma_f32(aWhh0[0][kk], ba, c0[0]);          // layer0(t+1) recurrence
            c0[1]  = wmma_f32(aWhh0[1][kk], ba, c0[1]);
        }
#pragma unroll
        for (int kk = 0; kk < 6; ++kk) {                         // layer0(t+1) input proj
            const int k0 = 4 * kk + 2 * half;
            v2f b;
            b[0] = xt[k0 < I_ ? k0 : I_ - 1];
            b[1] = xt[(k0 + 1) < I_ ? (k0 + 1) : I_ - 1];
            c0[0] = wmma_f32(aWih0[0][kk], b, c0[0]);
            c0[1] = wmma_f32(aWih0[1][kk], b, c0[1]);
        }

        // commit layer1: tanh -> lds[1], stream r_out (full 128B line per (b,t) row)
        float* ro = r_out + ((size_t)(b0 + n) * T_ + t) * H_;
#pragma unroll
        for (int mt = 0; mt < 2; ++mt) {
            float v[8];
#pragma unroll
            for (int r = 0; r < 8; ++r) {
                v[r] = fast_tanh(c1i[mt][r] + c1h[mt][r]);
                lds[1][16 * mt + 8 * half + r][n] = v[r];
            }
            const int m0 = 16 * mt + 8 * half;
            v4f p0 = {v[0], v[1], v[2], v[3]};
            v4f p1 = {v[4], v[5], v[6], v[7]};
            *(v4f*)(ro + m0)     = p0;
            *(v4f*)(ro + m0 + 4) = p1;
        }

        // commit layer0 (S0^{t+2}); skipped on last iteration so lds[0] ends as hT0
        if (!last) {
#pragma unroll
            for (int mt = 0; mt < 2; ++mt)
#pragma unroll
                for (int r = 0; r < 8; ++r)
                    lds[0][16 * mt + 8 * half + r][n] = fast_tanh(c0[mt][r]);
        }
    }

    // ================= epilogue: hT0/hT1 and output head ============================
#pragma unroll
    for (int l = 0; l < 2; ++l) {
        float* hp = hstate_out + ((size_t)l * B_ + (b0 + n)) * H_;
#pragma unroll
        for (int mt = 0; mt < 2; ++mt) {
            const int m0 = 16 * mt + 8 * half;
            v4f p0, p1;
#pragma unroll
            for (int q = 0; q < 4; ++q) {
                p0[q] = lds[l][m0 + q][n];
                p1[q] = lds[l][m0 + 4 + q][n];
            }
            *(v4f*)(hp + m0)     = p0;
            *(v4f*)(hp + m0 + 4) = p1;
        }
    }

    // outs[b] = sum_h S1_final[h][b] * Wout[h] + bout  (h-range split across halves)
    float partial = 0.0f;
#pragma unroll
    for (int i = 0; i < 16; ++i) {
        const int h = 16 * half + i;
        partial += lds[1][h][n] * Wout[h];
    }
    plds[lane] = partial;
    if (lane < 16)
        outs[b0 + n] = plds[n] + plds[n + 16] + bout[0];
}

extern "C" void kernel_launch(void* const* d_in, const int* in_sizes, int n_in,
                              void* d_out, int out_size, void* d_ws, size_t ws_size,
                              hipStream_t stream) {
    (void)in_sizes; (void)n_in; (void)d_ws; (void)ws_size; (void)out_size;
    const float* x     = (const float*)d_in[0];
    const float* h_in  = (const float*)d_in[1];
    const float* Wih0  = (const float*)d_in[2];
    const float* Whh0  = (const float*)d_in[3];
    const float* bih0  = (const float*)d_in[4];
    const float* bhh0  = (const float*)d_in[5];
    const float* Wih1  = (const float*)d_in[6];
    const float* Whh1  = (const float*)d_in[7];
    const float* bih1  = (const float*)d_in[8];
    const float* bhh1  = (const float*)d_in[9];
    const float* Wout  = (const float*)d_in[10];
    const float* bout  = (const float*)d_in[11];

    float* r_out  = (float*)d_out;                       // [B,T,H]
    float* outs   = r_out + (size_t)B_ * T_ * H_;        // [B]
    float* hstate = outs + B_;                           // [2,B,H]

    dim3 grid(B_ / 16);   // 64 independent batch tiles, one wave32 each
    dim3 block(32);
    rnn_fused_kernel<<<grid, block, 0, stream>>>(
        x, h_in, Wih0, Whh0, bih0, bhh0, Wih1, Whh1, bih1, bhh1, Wout, bout,
        r_out, outs, hstate);
}